// HHGR_41205916237976
// MI455X (gfx1250) — compile-verified
//
#include <hip/hip_runtime.h>
#include <math.h>

typedef __attribute__((ext_vector_type(2))) float v2f;
typedef __attribute__((ext_vector_type(4))) float v4f;
typedef __attribute__((ext_vector_type(8))) float v8f;

#define B_TOT 8192
#define G 32
#define D 64
#define WPB 4                 // waves (batch elements) per block
#define THREADS (WPB * 32)

__device__ __forceinline__ void wavebar() { __builtin_amdgcn_wave_barrier(); }

__global__ __launch_bounds__(THREADS) void hhgr_fused_kernel(
    const int*   __restrict__ group_inputs,   // [B]
    const int*   __restrict__ item_inputs,    // [B]
    const int*   __restrict__ members,        // [B,G]
    const float* __restrict__ user_table,     // [N_USERS,D]
    const float* __restrict__ item_table,     // [N_ITEMS,D]
    const float* __restrict__ group_table,    // [N_GROUPS,D]
    const float* __restrict__ group_embedds,  // [N_GROUPS,D]
    const float* __restrict__ att_w1,         // [128,16]
    const float* __restrict__ att_b1,         // [16]
    const float* __restrict__ att_w2,         // [16,1]
    const float* __restrict__ att_b2,         // [1]
    const float* __restrict__ cls_w,          // [1,2]
    const float* __restrict__ cls_b,          // [2]
    const float* __restrict__ pred_w1,        // [192,8]
    const float* __restrict__ pred_b1,        // [8]
    const float* __restrict__ pred_w2,        // [8,1]
    const float* __restrict__ pred_b2,        // [1]
    float* __restrict__ out_ne,               // [B,192]
    float* __restrict__ out_y,                // [B]
    float* __restrict__ out_at,               // [B,G]
    float* __restrict__ out_cls)              // [B]
{
    // ---- LDS ----
    __shared__ __align__(16) v2f   w1s[32 * 32];       // att_w1 swizzled for WMMA B-layout (8 KB)
    __shared__ float b1s[16];
    __shared__ float w2s[16];
    __shared__ __align__(16) float pw1[192 * 8];       // pred_w1 (6 KB)
    __shared__ __align__(16) float memL[WPB][G * D];   // gathered member rows (32 KB)
    __shared__ __align__(16) float itemL[WPB][D];      // item row per wave
    __shared__ float atL[WPB][G];                      // scores -> attention weights

    const int tid  = threadIdx.x;
    const int wave = tid >> 5;
    const int lane = tid & 31;

    // ---- block-cooperative weight staging ----
    // B-matrix 4x16 f32 layout: VGPR0 = rows {k0, k0+2} (half-wave split), VGPR1 = rows {k0+1, k0+3}
    for (int i = tid; i < 32 * 32; i += THREADS) {
        int t = i >> 5, l = i & 31;
        int n  = l & 15;
        int kr = t * 4 + ((l >> 4) << 1);
        v2f w;
        w.x = att_w1[kr * 16 + n];
        w.y = att_w1[(kr + 1) * 16 + n];
        w1s[i] = w;
    }
    if (tid < 16) { b1s[tid] = att_b1[tid]; w2s[tid] = att_w2[tid]; }
    for (int i = tid; i < 192 * 8; i += THREADS) pw1[i] = pred_w1[i];
    __syncthreads();

    const int b = blockIdx.x * WPB + wave;

    // ---- gather: lane g pulls its member's 256B row into LDS ----
    const int mi = members[b * G + lane];
    const float* urow = user_table + (long)mi * D;
    #pragma unroll
    for (int c = 0; c < D; c += 4)
        *(v4f*)&memL[wave][lane * D + c] = *(const v4f*)&urow[c];

    const int ii = item_inputs[b];
    if (lane < 16)
        *(v4f*)&itemL[wave][lane * 4] = *(const v4f*)&item_table[(long)ii * D + lane * 4];
    wavebar();

    // ---- attention MLP via V_WMMA_F32_16X16X4_F32, K=128 in 32 steps ----
    {
        const int n    = lane & 15;
        const int koff = (lane >> 4) << 1;           // A-layout K half-offset
        const int m0   = lane & 15;                  // A row within tile
        v8f acc0 = {0.f,0.f,0.f,0.f,0.f,0.f,0.f,0.f};
        v8f acc1 = acc0;
        #pragma unroll
        for (int t = 0; t < 32; ++t) {
            const int k0 = t * 4;
            v2f bm = w1s[t * 32 + lane];
            v2f a0, a1;
            if (k0 < 64) {                           // member-embedding half of gi
                a0 = *(const v2f*)&memL[wave][m0 * D + k0 + koff];
                a1 = *(const v2f*)&memL[wave][(m0 + 16) * D + k0 + koff];
            } else {                                 // broadcast item half of gi
                a0 = *(const v2f*)&itemL[wave][k0 - 64 + koff];
                a1 = a0;
            }
            acc0 = __builtin_amdgcn_wmma_f32_16x16x4_f32(
                       false, a0, false, bm, (short)0, acc0, false, false);
            acc1 = __builtin_amdgcn_wmma_f32_16x16x4_f32(
                       false, a1, false, bm, (short)0, acc1, false, false);
        }

        // ---- ReLU + second linear + per-member score, from C layout ----
        const float b1v = b1s[n];
        const float w2v = w2s[n];
        const float b2v = att_b2[0];
        #pragma unroll
        for (int j = 0; j < 8; ++j) {
            float h0 = fmaxf(acc0[j] + b1v, 0.f) * w2v;
            float h1 = fmaxf(acc1[j] + b1v, 0.f) * w2v;
            #pragma unroll
            for (int m = 1; m <= 8; m <<= 1) {       // 16-lane half butterfly
                h0 += __shfl_xor(h0, m, 32);
                h1 += __shfl_xor(h1, m, 32);
            }
            if (n == 0) {
                int ml = j + ((lane >> 4) << 3);     // row M = j (+8 for upper half)
                atL[wave][ml]      = h0 + b2v;       // members 0..15
                atL[wave][16 + ml] = h1 + b2v;       // members 16..31
            }
        }
    }
    wavebar();

    // ---- softmax over members (wave32 shuffles) ----
    float sc = atL[wave][lane];
    float mx = sc;
    #pragma unroll
    for (int m = 1; m <= 16; m <<= 1) mx = fmaxf(mx, __shfl_xor(mx, m, 32));
    float e = __expf(sc - mx);
    float sum = e;
    #pragma unroll
    for (int m = 1; m <= 16; m <<= 1) sum += __shfl_xor(sum, m, 32);
    float at = e / sum;
    out_at[b * G + lane] = at;

    // ---- argmax (first-index tie-break) ----
    float v = at; int idx = lane;
    #pragma unroll
    for (int m = 1; m <= 16; m <<= 1) {
        float ov = __shfl_xor(v, m, 32);
        int   oi = __shfl_xor(idx, m, 32);
        if (ov > v || (ov == v && oi < idx)) { v = ov; idx = oi; }
    }

    // ---- routing classifier ----
    float s0 = v * cls_w[0] + cls_b[0];
    float s1 = v * cls_w[1] + cls_b[1] + 0.5f;       // CLS_BIAS
    int pcls = (s1 > s0) ? 1 : 0;
    if (lane == 0) out_cls[b] = (float)pcls;

    wavebar();
    atL[wave][lane] = at;                            // publish weights for weighted sum
    wavebar();

    // ---- g_att (leader or attention-weighted), each lane owns d=lane and d=lane+32 ----
    float ga0, ga1;
    if (pcls == 1) {
        ga0 = memL[wave][idx * D + lane];
        ga1 = memL[wave][idx * D + lane + 32];
    } else {
        ga0 = 0.f; ga1 = 0.f;
        #pragma unroll 4
        for (int g = 0; g < G; ++g) {
            float w = atL[wave][g];
            ga0 = fmaf(w, memL[wave][g * D + lane],      ga0);
            ga1 = fmaf(w, memL[wave][g * D + lane + 32], ga1);
        }
    }

    const int gidx = group_inputs[b];
    float gp0 = group_table[(long)gidx * D + lane]      + group_embedds[(long)gidx * D + lane];
    float gp1 = group_table[(long)gidx * D + lane + 32] + group_embedds[(long)gidx * D + lane + 32];
    float g0 = ga0 + gp0, g1 = ga1 + gp1;
    float it0 = itemL[wave][lane], it1 = itemL[wave][lane + 32];
    float e0 = g0 * it0, e1 = g1 * it1;

    float* ne = out_ne + (long)b * 192;
    ne[lane]        = e0;  ne[lane + 32]       = e1;
    ne[64 + lane]   = g0;  ne[96 + lane]       = g1;
    ne[128 + lane]  = it0; ne[160 + lane]      = it1;

    // ---- predict MLP: [192] -> 8 -> 1 ----
    float ph[8];
    #pragma unroll
    for (int j = 0; j < 8; ++j) {
        float p = e0 * pw1[lane * 8 + j]
                + e1 * pw1[(lane + 32) * 8 + j]
                + g0 * pw1[(64 + lane) * 8 + j]
                + g1 * pw1[(96 + lane) * 8 + j]
                + it0 * pw1[(128 + lane) * 8 + j]
                + it1 * pw1[(160 + lane) * 8 + j];
        #pragma unroll
        for (int m = 1; m <= 16; m <<= 1) p += __shfl_xor(p, m, 32);
        ph[j] = p;
    }
    if (lane == 0) {
        float acc = pred_b2[0];
        #pragma unroll
        for (int j = 0; j < 8; ++j)
            acc = fmaf(fmaxf(ph[j] + pred_b1[j], 0.f), pred_w2[j], acc);
        out_y[b] = 1.f / (1.f + __expf(-acc));
    }
}

extern "C" void kernel_launch(void* const* d_in, const int* in_sizes, int n_in,
                              void* d_out, int out_size, void* d_ws, size_t ws_size,
                              hipStream_t stream) {
    (void)in_sizes; (void)n_in; (void)d_ws; (void)ws_size; (void)out_size;

    const int*   group_inputs  = (const int*)d_in[0];
    const int*   item_inputs   = (const int*)d_in[1];
    const int*   members       = (const int*)d_in[2];
    const float* user_table    = (const float*)d_in[3];
    const float* item_table    = (const float*)d_in[4];
    const float* group_table   = (const float*)d_in[5];
    const float* group_embedds = (const float*)d_in[6];
    const float* att_w1        = (const float*)d_in[7];
    const float* att_b1        = (const float*)d_in[8];
    const float* att_w2        = (const float*)d_in[9];
    const float* att_b2        = (const float*)d_in[10];
    const float* cls_w         = (const float*)d_in[11];
    const float* cls_b         = (const float*)d_in[12];
    const float* pred_w1       = (const float*)d_in[13];
    const float* pred_b1       = (const float*)d_in[14];
    const float* pred_w2       = (const float*)d_in[15];
    const float* pred_b2       = (const float*)d_in[16];

    float* out = (float*)d_out;
    float* out_ne  = out;                          // 8192*192
    float* out_y   = out_ne + (long)B_TOT * 192;   // 8192
    float* out_at  = out_y  + B_TOT;               // 8192*32
    float* out_cls = out_at + (long)B_TOT * G;     // 8192

    dim3 grid(B_TOT / WPB);
    dim3 block(THREADS);
    hhgr_fused_kernel<<<grid, block, 0, stream>>>(
        group_inputs, item_inputs, members, user_table, item_table,
        group_table, group_embedds, att_w1, att_b1, att_w2, att_b2,
        cls_w, cls_b, pred_w1, pred_b1, pred_w2, pred_b2,
        out_ne, out_y, out_at, out_cls);
}